// KineticPooling_57131654972039
// MI455X (gfx1250) — compile-verified
//
#include <hip/hip_runtime.h>
#include <hip/hip_bf16.h>

typedef __attribute__((ext_vector_type(2))) float v2f;
typedef __attribute__((ext_vector_type(8))) float v8f;

#define NMO   32     // molecular orbitals (columns of each MO slice)
#define NROWS 16     // electrons total (8 up + 8 down)
#define CPB   16     // configs handled per block (8 waves x 2 configs)
#define TPB   256

// ds_swizzle xor within the 32-lane wave: srcLane = lane ^ MASK.
// Immediate pattern, no index VGPR, single DS op.
template <int MASK>
__device__ __forceinline__ float swz_xor(float x) {
  return __int_as_float(
      __builtin_amdgcn_ds_swizzle(__float_as_int(x), (MASK << 10) | 0x1f));
}

// Sum within aligned groups of 8 lanes; every lane of a group gets the sum.
__device__ __forceinline__ float red8(float v) {
  v += swz_xor<1>(v);
  v += swz_xor<2>(v);
  v += swz_xor<4>(v);
  return v;
}

// Per-problem trace of a block-diagonal 16x16 in WMMA C-layout.
// fm[] is a per-lane one-hot over the 8 C registers: 1.0 at the diagonal slot
// this lane owns (lane j -> reg j for P0, lane j+24 -> reg j for P1), else 0.
// Masked dot + group-of-8 reduce; result is lane-local (lanes 0-7 -> tr(P0),
// lanes 24-31 -> tr(P1)). Pure VALU tree: no dynamic vector extract, which
// the compiler would otherwise lower through an LDS round-trip.
__device__ __forceinline__ float trace_local(const float* p, const float* fm) {
  const float s0 = fm[0] * p[0] + fm[1] * p[1];
  const float s1 = fm[2] * p[2] + fm[3] * p[3];
  const float s2 = fm[4] * p[4] + fm[5] * p[5];
  const float s3 = fm[6] * p[6] + fm[7] * p[7];
  return red8((s0 + s1) + (s2 + s3));
}

// C-layout -> B-layout (4x16 K-chunks) needs only identity or lane^16 moves:
// B chunk q, element e, lane l wants N[4q+e+2*(l>=16)][l&15]; the source lane
// is l (same half) or l^16 (other half). sw[] holds the xor-16 swap of nC[].
__device__ __forceinline__ void conv_b(const float* nC, const float* sw,
                                       bool lo, v2f* bq) {
#pragma unroll
  for (int e = 0; e < 2; ++e) {
    bq[0][e] = lo ? nC[0 + e] : sw[2 + e];
    bq[1][e] = lo ? nC[4 + e] : sw[6 + e];
    bq[2][e] = lo ? sw[0 + e] : nC[2 + e];
    bq[3][e] = lo ? sw[4 + e] : nC[6 + e];
  }
}

// Faddeev-LeVerrier on two block-diagonal 8x8 problems per wave (wave32).
//   A_k = A * B_{k-1};  c_k = -tr(A_k)/k;  B_k = A_k + c_k I
//   det(A) = c_8 (n=8 even);  adj(A) = -B_7;  det*tr(A^-1 Bm) = -tr(Bm * B_7)
// Results are LANE-LOCAL: lanes 0-7 hold problem 0, lanes 24-31 problem 1.
__device__ __forceinline__ void spin_flv(const float* __restrict__ sA,
                                         const float* __restrict__ sB,
                                         const int* __restrict__ idx0,
                                         const int* __restrict__ idx1,
                                         int lane, const float* fm,
                                         float& detL, float& TL) {
  const int  m     = lane & 15;        // M coordinate in A-layout
  const int  khalf = lane >> 4;        // K half select in A-layout
  const int  prow  = m >> 3;           // which 8x8 problem owns this row
  const int  rowA  = m & 7;
  const int* idxm  = prow ? idx1 : idx0;
  const bool lo    = (lane < 16);

  // A in WMMA A-layout chunks: chunk q, element e holds K=4q+2*khalf+e, M=lane&15
  v2f aA[4];
#pragma unroll
  for (int q = 0; q < 4; ++q)
#pragma unroll
    for (int e = 0; e < 2; ++e) {
      const int k = 4 * q + 2 * khalf + e;
      aA[q][e] = ((k >> 3) == prow) ? sA[rowA * NMO + idxm[k & 7]] : 0.f;
    }

  // A in C-layout to seed the recursion (block-diagonal 16x16):
  // reg j: lanes 0-7 -> P0[j][lane], lanes 24-31 -> P1[j][lane-24], else 0.
  float nC[8];
  {
    const bool act = (lane < 8) || (lane >= 24);
    const int* idn = (lane >= 24) ? idx1 : idx0;
    const int  col = idn[lane & 7];
#pragma unroll
    for (int j = 0; j < 8; ++j)
      nC[j] = act ? sA[j * NMO + col] : 0.f;
  }

  // c_1 = -tr(A); B_1 = A + c_1 I   (c stays lane-local per problem)
  float ck = -trace_local(nC, fm);
#pragma unroll
  for (int j = 0; j < 8; ++j) nC[j] = fmaf(fm[j], ck, nC[j]);

  float b7[8];
  detL = 0.f;
#pragma unroll
  for (int k = 2; k <= 8; ++k) {
    // P = A * N : one 16x16x16 f32 product = 4 chained WMMA 16x16x4
    float sw[8];
#pragma unroll
    for (int j = 0; j < 8; ++j) sw[j] = swz_xor<16>(nC[j]);
    v2f bq[4];
    conv_b(nC, sw, lo, bq);

    v8f acc = {0.f, 0.f, 0.f, 0.f, 0.f, 0.f, 0.f, 0.f};
#pragma unroll
    for (int q = 0; q < 4; ++q)
      acc = __builtin_amdgcn_wmma_f32_16x16x4_f32(false, aA[q], false, bq[q],
                                                  (short)0, acc, false, false);
    float pv[8];
#pragma unroll
    for (int j = 0; j < 8; ++j) pv[j] = acc[j];

    ck = -trace_local(pv, fm) * (1.0f / (float)k);
    if (k == 8) {
      detL = ck;                       // det(A) = c_8
    } else {
#pragma unroll
      for (int j = 0; j < 8; ++j) {
        nC[j] = fmaf(fm[j], ck, pv[j]);
        if (k == 7) b7[j] = nC[j];     // B_7; adj(A) = -B_7
      }
    }
  }

  // T = tr(adj(A)*Bm) = -tr(Bm * B_7): left = Bm gathered in A-layout,
  // right = B_7 re-striped from C-layout.
  v2f bmA[4];
#pragma unroll
  for (int q = 0; q < 4; ++q)
#pragma unroll
    for (int e = 0; e < 2; ++e) {
      const int k = 4 * q + 2 * khalf + e;
      bmA[q][e] = ((k >> 3) == prow) ? sB[rowA * NMO + idxm[k & 7]] : 0.f;
    }

  float sw[8];
#pragma unroll
  for (int j = 0; j < 8; ++j) sw[j] = swz_xor<16>(b7[j]);
  v2f bq[4];
  conv_b(b7, sw, lo, bq);

  v8f acc = {0.f, 0.f, 0.f, 0.f, 0.f, 0.f, 0.f, 0.f};
#pragma unroll
  for (int q = 0; q < 4; ++q)
    acc = __builtin_amdgcn_wmma_f32_16x16x4_f32(false, bmA[q], false, bq[q],
                                                (short)0, acc, false, false);
  float pv[8];
#pragma unroll
  for (int j = 0; j < 8; ++j) pv[j] = acc[j];

  TL = -trace_local(pv, fm);
}

__global__ __launch_bounds__(TPB)
void kinetic_pool_flv(const float* __restrict__ MO,
                      const float* __restrict__ d2MO,
                      const int* __restrict__ cup,
                      const int* __restrict__ cdown,
                      float* __restrict__ out,
                      int nbatch) {
  __shared__ float sMO[NROWS * NMO];
  __shared__ float sD2[NROWS * NMO];
  __shared__ int   sUp[CPB][8];
  __shared__ int   sDn[CPB][8];

  const int b     = blockIdx.y;
  const int cBase = blockIdx.x * CPB;
  const int tid   = threadIdx.x;

  for (int i = tid; i < NROWS * NMO; i += TPB) {
    sMO[i] = MO[(size_t)b * (NROWS * NMO) + i];
    sD2[i] = d2MO[(size_t)b * (NROWS * NMO) + i];
  }
  for (int i = tid; i < CPB * 8; i += TPB) {
    sUp[i >> 3][i & 7] = cup[(size_t)(cBase + (i >> 3)) * 8 + (i & 7)];
    sDn[i >> 3][i & 7] = cdown[(size_t)(cBase + (i >> 3)) * 8 + (i & 7)];
  }
  __syncthreads();

  const int wid  = tid >> 5;      // 8 waves per block
  const int lane = tid & 31;
  const int c0l  = 2 * wid;       // two configs per wave (block-diag packing)
  const int c1l  = c0l + 1;

  // One-hot diagonal mask: reg j's diagonal lives at lanes j (P0), j+24 (P1).
  float fm[8];
#pragma unroll
  for (int j = 0; j < 8; ++j)
    fm[j] = ((lane == j) || (lane == j + 24)) ? 1.0f : 0.0f;

  // Lane-local results: lanes 0-7 carry config c0, lanes 24-31 carry c1.
  float dU, TU, dD, TD;
  spin_flv(sMO,           sD2,           sUp[c0l], sUp[c1l], lane, fm, dU, TU);
  spin_flv(sMO + 8 * NMO, sD2 + 8 * NMO, sDn[c0l], sDn[c1l], lane, fm, dD, TD);

  // out[c][b] = -0.5 * (tr_up + tr_dn) * detU * detD
  //           = -0.5 * (T_up * detD + T_dn * detU), T_s = det_s*tr(A_s^-1 B_s)
  const float o = -0.5f * (TU * dD + TD * dU);
  if (lane == 0)  out[(size_t)(cBase + c0l) * nbatch + b] = o;
  if (lane == 24) out[(size_t)(cBase + c1l) * nbatch + b] = o;
}

extern "C" void kernel_launch(void* const* d_in, const int* in_sizes, int n_in,
                              void* d_out, int out_size, void* d_ws, size_t ws_size,
                              hipStream_t stream) {
  const float* MO   = (const float*)d_in[0];
  const float* d2MO = (const float*)d_in[1];
  const int*   cup  = (const int*)d_in[2];
  const int*   cdn  = (const int*)d_in[3];
  float*       out  = (float*)d_out;

  const int nbatch = in_sizes[0] / (NROWS * NMO);  // 8192
  const int nconf  = in_sizes[2] / 8;              // 64

  dim3 grid(nconf / CPB, nbatch);                  // (4, 8192)
  kinetic_pool_flv<<<grid, TPB, 0, stream>>>(MO, d2MO, cup, cdn, out, nbatch);

  (void)n_in; (void)out_size; (void)d_ws; (void)ws_size;
}